// RegLoss_36172214567283
// MI455X (gfx1250) — compile-verified
//
#include <hip/hip_runtime.h>
#include <hip/hip_bf16.h>

typedef __attribute__((ext_vector_type(2))) float v2f;
typedef __attribute__((ext_vector_type(8))) float v8f;

#define B_   128
#define J_   17
#define Y_   96
#define X_   72
#define YX_  (Y_ * X_)      // 6912
#define NQ_  (B_ * J_)      // 2176 (b*J + j) slices
#define NV4_ (YX_ / 4)      // 1728 float4 per slice per channel
#define NTHREADS_ 256

// Full-wave (32 lane) sum via V_WMMA_F32_16X16X4_F32.
// A layout (16x4 f32): lanes 0-15 hold A[m][0],A[m][1]; lanes 16-31 hold A[m][2],A[m][3].
// With A[.][0]=partial, A[.][1]=0 and B=ones, D[m][n] = p_m + p_{m+16} for every n.
// Summing the 8 D VGPRs per lane gives half-sums; one shfl_xor(16) completes the wave sum.
// NOTE: EXEC must be all 1s at call sites (ISA 7.12 restriction) — callers guarantee it.
__device__ __forceinline__ float wave_sum_wmma(float v) {
    v2f a; a[0] = v;    a[1] = 0.0f;
    v2f o; o[0] = 1.0f; o[1] = 1.0f;
    v8f c = {};
    c = __builtin_amdgcn_wmma_f32_16x16x4_f32(
        /*neg_a=*/false, a, /*neg_b=*/false, o,
        /*c_mod=*/(short)0, c, /*reuse_a=*/false, /*reuse_b=*/false);
    float s = ((c[0] + c[1]) + (c[2] + c[3])) + ((c[4] + c[5]) + (c[6] + c[7]));
    s += __shfl_xor(s, 16, 32);   // wave32: lanes 0-15 have M=0..7 half, 16-31 have M=8..15 half
    return s;                      // every lane: full 32-lane sum
}

__device__ __forceinline__ float block_sum(float v, float* lds8) {
    float s = wave_sum_wmma(v);
    const int tid = threadIdx.x;
    if ((tid & 31) == 0) lds8[tid >> 5] = s;
    __syncthreads();
    float t = 0.0f;
    if (tid == 0) {
        #pragma unroll
        for (int k = 0; k < NTHREADS_ / 32; ++k) t += lds8[k];
    }
    return t;  // valid on tid==0 only
}

// One block per q = b*J + j slice. Streams 6912 floats from each channel of loc
// as float4 (slice base is 27648 B => 16B aligned), accumulates squared errors.
__global__ void __launch_bounds__(NTHREADS_)
regloss_partial_kernel(const float* __restrict__ loc,
                       const float* __restrict__ cord,
                       const float* __restrict__ tw,
                       float* __restrict__ ws) {
    const int q   = blockIdx.x;        // b*J + j
    const int tid = threadIdx.x;
    // Faithful reshape-bug indexing: gt uses jj = q/B, bb = q%B into cord (2,J,B).
    const int jj = q >> 7;             // / 128
    const int bb = q & 127;            // % 128
    const float w  = tw[q];            // target_weight flat (B,J,1) == index q
    const float cx = cord[jj * B_ + bb];
    const float cy = cord[J_ * B_ + jj * B_ + bb];

    const float4* __restrict__ p0 =
        (const float4*)(loc + (size_t)q * YX_);                       // channel x
    const float4* __restrict__ p1 =
        (const float4*)(loc + (size_t)NQ_ * YX_ + (size_t)q * YX_);   // channel y

    float acc = 0.0f;
    for (int i = tid; i < NV4_; i += NTHREADS_) {
        const float4 a = p0[i];
        const float4 b = p1[i];
        const int o  = i << 2;         // X_=72 is a multiple of 4: float4 never crosses a row
        const int y  = o / X_;
        const int x0 = o - y * X_;
        const float gyw = truncf((float)y - cy) * w;   // same y for all 4 lanes of the vec

        float gx, dx, dy;
        gx = truncf((float)(x0 + 0) - cx) * w; dx = a.x * w - gx; dy = b.x * w - gyw;
        acc = fmaf(dx, dx, acc); acc = fmaf(dy, dy, acc);
        gx = truncf((float)(x0 + 1) - cx) * w; dx = a.y * w - gx; dy = b.y * w - gyw;
        acc = fmaf(dx, dx, acc); acc = fmaf(dy, dy, acc);
        gx = truncf((float)(x0 + 2) - cx) * w; dx = a.z * w - gx; dy = b.z * w - gyw;
        acc = fmaf(dx, dx, acc); acc = fmaf(dy, dy, acc);
        gx = truncf((float)(x0 + 3) - cx) * w; dx = a.w * w - gx; dy = b.w * w - gyw;
        acc = fmaf(dx, dx, acc); acc = fmaf(dy, dy, acc);
    }

    __shared__ float lds8[NTHREADS_ / 32];
    const float t = block_sum(acc, lds8);
    if (tid == 0) ws[q] = t;
}

// Single block: reduce the 2176 per-slice sums, apply 0.5/(J*B*YX), write scalar.
__global__ void __launch_bounds__(NTHREADS_)
regloss_final_kernel(const float* __restrict__ ws, float* __restrict__ out) {
    const int tid = threadIdx.x;
    float acc = 0.0f;
    for (int i = tid; i < NQ_; i += NTHREADS_) acc += ws[i];

    __shared__ float lds8[NTHREADS_ / 32];
    const float t = block_sum(acc, lds8);
    if (tid == 0) {
        const float scale = 0.5f / (float)((long long)J_ * B_ * YX_);  // 0.5 / 15040512
        out[0] = t * scale;
    }
}

extern "C" void kernel_launch(void* const* d_in, const int* in_sizes, int n_in,
                              void* d_out, int out_size, void* d_ws, size_t ws_size,
                              hipStream_t stream) {
    (void)in_sizes; (void)n_in; (void)out_size; (void)ws_size;
    const float* loc  = (const float*)d_in[0];  // (2, B, J, Y, X) f32
    const float* cord = (const float*)d_in[1];  // (2, J, B) f32
    const float* tw   = (const float*)d_in[2];  // (B, J, 1) f32
    float* out = (float*)d_out;                 // scalar f32
    float* ws  = (float*)d_ws;                  // NQ_ floats of scratch

    regloss_partial_kernel<<<NQ_, NTHREADS_, 0, stream>>>(loc, cord, tw, ws);
    regloss_final_kernel<<<1, NTHREADS_, 0, stream>>>(ws, out);
}